// ROIAlign_25099788877848
// MI455X (gfx1250) — compile-verified
//
#include <hip/hip_runtime.h>

typedef __attribute__((ext_vector_type(2))) float v2f;
typedef __attribute__((ext_vector_type(8))) float v8f;

#define RA_C 256
#define RA_H 200
#define RA_W 304
#define RA_PH 7
#define RA_PW 7
#define RA_SCALE 0.25f
#define RA_PLANE (RA_H * RA_W)

// One block per ROI, 512 threads = 16 wave32s, one 16-channel block per wave.
// Per pooled row ph: batch-gather 2x28 taps (32-bit GVS addressing off the
// uniform feat base), y-interp into B, then 14 chained V_WMMA_F32_16X16X4_F32
// (K=28 x-taps x 2 sample rows). 2x2 grid pooled with in-lane acc-pair adds.
__global__ __launch_bounds__(512) void roi_align_wmma(
    const float* __restrict__ feat, const float* __restrict__ rois,
    float* __restrict__ out)
{
  __shared__ float s_w0x[16], s_w1x[16], s_wy0[16], s_wy1[16];
  __shared__ int   s_xl[16], s_xh[16], s_yl[16], s_yh[16];
  __shared__ int   s_b;

  const int r   = blockIdx.x;
  const int tid = threadIdx.x;

  // ---- per-ROI tap tables (lanes 0..15 -> x axis, lanes 16..31 -> y axis) ----
  if (tid < 32) {
    const float* roi = rois + (size_t)r * 5;
    const float px1 = roi[1] * RA_SCALE - 0.5f;
    const float py1 = roi[2] * RA_SCALE - 0.5f;
    const float px2 = roi[3] * RA_SCALE - 0.5f;
    const float py2 = roi[4] * RA_SCALE - 0.5f;
    const float bw  = (px2 - px1) * (1.0f / RA_PW);
    const float bh  = (py2 - py1) * (1.0f / RA_PH);
    if (tid == 0) s_b = (int)roi[0];

    const int  i   = tid & 15;
    const bool isY = tid >= 16;
    if (i < 14) {
      const float start = isY ? py1 : px1;
      const float bin   = isY ? bh  : bw;
      const int   size  = isY ? RA_H : RA_W;
      // sample coord: i//2 + (i%2 + 0.5)/2
      const float coord = start + ((float)(i >> 1) + (float)(i & 1) * 0.5f + 0.25f) * bin;
      const bool  valid = (coord > -1.0f) && (coord < (float)size);
      const float c0v = fmaxf(coord, 0.0f);
      const float fl  = floorf(c0v);
      const int   low  = (int)fminf(fl, (float)(size - 1));
      const int   high = min(low + 1, size - 1);
      const float frac = (fl >= (float)(size - 1)) ? 0.0f : (c0v - (float)low);
      const float w0 = valid ? (1.0f - frac) : 0.0f;   // validity folded into weights
      const float w1 = valid ? frac          : 0.0f;
      if (isY) { s_yl[i] = low; s_yh[i] = high; s_wy0[i] = w0; s_wy1[i] = w1; }
      else     { s_xl[i] = low; s_xh[i] = high; s_w0x[i] = w0; s_w1x[i] = w1; }
    } else {
      if (isY) { s_yl[i] = 0; s_yh[i] = 0; s_wy0[i] = 0.0f; s_wy1[i] = 0.0f; }
      else     { s_xl[i] = 0; s_xh[i] = 0; s_w0x[i] = 0.0f; s_w1x[i] = 0.0f; }
    }
  }
  __syncthreads();

  const int  lane = tid & 31;
  const int  cb   = tid >> 5;      // channel block 0..15
  const int  m    = lane & 15;     // M row (sample-x index) / channel-in-block
  const bool hi   = lane >= 16;    // upper half-wave carries K-rows +2 / M-rows +8
  const int  ch   = cb * 16 + m;

  // 32-bit element offset of this lane's channel plane (124 MB tensor < 2^31 B,
  // so all gathers use uniform-SGPR-base + 32-bit VGPR index addressing).
  const unsigned planeOff = (unsigned)(s_b * RA_C + ch) * (unsigned)RA_PLANE;

  // A matrix (16x28 x-tap weight matrix, block-diagonal): built once per ROI.
  // Chunk t covers K = 4t..4t+3. VGPR0 -> K=4t (lanes0-15) / 4t+2 (lanes16-31),
  // VGPR1 -> K=4t+1 / 4t+3. Even K = xl-tap weight, odd K = xh-tap weight.
  float    a0[7], a1[7];
  unsigned baseA[7], baseB[7];   // planeOff + x-tap column, loop-invariant
#pragma unroll
  for (int t = 0; t < 7; ++t) {
    const int  s   = 2 * t + (hi ? 1 : 0);   // sample-x index of this lane's K pair
    const bool hit = (m == s);
    a0[t] = hit ? s_w0x[s] : 0.0f;
    a1[t] = hit ? s_w1x[s] : 0.0f;
    baseA[t] = planeOff + (unsigned)s_xl[s];
    baseB[t] = planeOff + (unsigned)s_xh[s];
  }

  for (int ph = 0; ph < RA_PH; ++ph) {
    v8f acc = {};
#pragma unroll
    for (int iy = 0; iy < 2; ++iy) {
      const int      sy   = 2 * ph + iy;
      const unsigned rowL = (unsigned)(s_yl[sy] * RA_W);
      const unsigned rowH = (unsigned)(s_yh[sy] * RA_W);
      const float    wy0  = s_wy0[sy];
      const float    wy1  = s_wy1[sy];

      // Batch all 28 gathers for this sample row first (load clause, latency hide).
      float fA0[7], fA1[7], fB0[7], fB1[7];
#pragma unroll
      for (int t = 0; t < 7; ++t) {
        fA0[t] = feat[baseA[t] + rowL];
        fA1[t] = feat[baseA[t] + rowH];
        fB0[t] = feat[baseB[t] + rowL];
        fB1[t] = feat[baseB[t] + rowH];
      }
      // y-interp into B rows, then chained k-steps.
#pragma unroll
      for (int t = 0; t < 7; ++t) {
        v2f A; A.x = a0[t]; A.y = a1[t];
        v2f B;
        B.x = fmaf(wy1, fA1[t], wy0 * fA0[t]);
        B.y = fmaf(wy1, fB1[t], wy0 * fB0[t]);
        acc = __builtin_amdgcn_wmma_f32_16x16x4_f32(
            false, A, false, B, (short)0, acc, false, false);
      }
    }
    // D layout: VGPR v = M=v (lanes0-15) / M=v+8 (lanes16-31); pool sx pairs in-lane.
    const float p0 = (acc[0] + acc[1]) * 0.25f;  // pw 0 / 4
    const float p1 = (acc[2] + acc[3]) * 0.25f;  // pw 1 / 5
    const float p2 = (acc[4] + acc[5]) * 0.25f;  // pw 2 / 6
    const float p3 = (acc[6] + acc[7]) * 0.25f;  // pw 3 / (7 = pad, dropped)

    const unsigned oidx = ((unsigned)r * RA_C + (unsigned)ch) * (RA_PH * RA_PW)
                        + (unsigned)(ph * RA_PW) + (hi ? 4u : 0u);
    out[oidx + 0] = p0;
    out[oidx + 1] = p1;
    out[oidx + 2] = p2;
    if (!hi) out[oidx + 3] = p3;
  }
}

extern "C" void kernel_launch(void* const* d_in, const int* in_sizes, int n_in,
                              void* d_out, int out_size, void* d_ws, size_t ws_size,
                              hipStream_t stream) {
  const float* feat = (const float*)d_in[0];
  const float* rois = (const float*)d_in[1];
  float* out = (float*)d_out;
  const int R = in_sizes[1] / 5;   // 1024
  roi_align_wmma<<<dim3(R), dim3(512), 0, stream>>>(feat, rois, out);
}